// Encoder_37598143709729
// MI455X (gfx1250) — compile-verified
//
#include <hip/hip_runtime.h>

typedef float v2f __attribute__((ext_vector_type(2)));
typedef float v8f __attribute__((ext_vector_type(8)));

#define IN_CH 256
#define HID   128
#define KTILE 64

// ---------------- degree / norm precompute ----------------

// deg starts at 1.0 (self loop contributes 1 to every destination)
__global__ void k_deg_init(float* __restrict__ deg, int n) {
    int i = blockIdx.x * blockDim.x + threadIdx.x;
    if (i < n) deg[i] = 1.0f;
}

// one thread per edge: deg[col[e]] += 1
__global__ void k_deg_edges(const int* __restrict__ ei, float* __restrict__ deg, int E) {
    int e = blockIdx.x * blockDim.x + threadIdx.x;
    if (e < E) atomicAdd(&deg[ei[(size_t)E + e]], 1.0f);
}

// dinv = rsqrt(deg), in place (deg >= 1 always, so no zero guard needed)
__global__ void k_dinv(float* __restrict__ deg, int n) {
    int i = blockIdx.x * blockDim.x + threadIdx.x;
    if (i < n) deg[i] = rsqrtf(deg[i]);
}

// ---------------- fp32 WMMA GEMM: h = x @ W, fused self-loop + bias epilogue ----------------
// Each wave: one 16-row strip (M tile), all 8 N tiles (128 cols), K swept in steps of 4.
// Block of 8 waves stages W K-chunks in LDS using a K-pair interleaved layout:
//   sW[pair * (2*HID) + n*2 + j] = W[kb + 2*pair + j][n]
// so a B fragment for (k-pair, n) is one aligned 8-byte ds_load_b64 -> even VGPR pair.
__global__ __launch_bounds__(256) void k_gemm_wmma(
    const float* __restrict__ x, const float* __restrict__ W,
    const float* __restrict__ bias, const float* __restrict__ dinv,
    float* __restrict__ h, float* __restrict__ out, int nNodes) {
    __shared__ float sW[KTILE * HID];   // 32 KB

    const int tid  = threadIdx.x;
    const int wave = tid >> 5;
    const int lane = tid & 31;
    const int half = lane >> 4;   // 0: lanes 0-15 (K=kl,kl+1 / M=0..7), 1: lanes 16-31 (K=kl+2,kl+3 / M=8..15)
    const int r    = lane & 15;
    const int m0   = (blockIdx.x * 8 + wave) * 16;
    const bool act = (m0 < nNodes);   // N is a multiple of 16 -> full strips only

    v8f acc[8] = {};   // 8 N-tiles of 16x16 f32 accumulators (64 VGPRs)

    for (int kb = 0; kb < IN_CH; kb += KTILE) {
        // prefetch next W chunk (32 KB; 256 threads x 128 B)
        if (kb + KTILE < IN_CH)
            __builtin_prefetch(W + (size_t)(kb + KTILE) * HID + tid * 32, 0, 1);

        __syncthreads();
        // cooperative interleaved stage of W[kb:kb+64, :]
#pragma unroll
        for (int p = 0; p < (KTILE * HID / 2) / 256; ++p) {   // 16 pairs per thread
            const int i  = tid + 256 * p;     // flat pair index, 0..4095
            const int kp = i >> 7;            // K pair 0..31
            const int n  = i & 127;           // column
            float2 w2;
            w2.x = W[(size_t)(kb + 2 * kp) * HID + n];
            w2.y = W[(size_t)(kb + 2 * kp + 1) * HID + n];
            *(float2*)(sW + (size_t)i * 2) = w2;
        }
        __syncthreads();

        if (act) {
            // A fragment source row for this lane (8B-aligned: kb, kl, 2*half all even)
            const float* xp = x + (size_t)(m0 + r) * IN_CH + kb + 2 * half;
#pragma unroll
            for (int kl = 0; kl < KTILE; kl += 4) {
                v2f a = *(const v2f*)(xp + kl);                       // global_load_b64
                const float* wrow = sW + ((kl >> 1) + half) * (2 * HID);
#pragma unroll
                for (int nt = 0; nt < 8; ++nt) {
                    v2f b = *(const v2f*)(wrow + (nt * 16 + r) * 2);  // ds_load_b64
                    acc[nt] = __builtin_amdgcn_wmma_f32_16x16x4_f32(
                        false, a, false, b, (short)0, acc[nt], false, false);
                }
            }
        }
    }

    if (act) {
        // epilogue: h[m][n] = acc ; out[m][n] = bias[n] + dinv[m]^2 * acc   (self-loop msg + bias)
        float d2[8];
#pragma unroll
        for (int i = 0; i < 8; ++i) {
            float dv = dinv[m0 + i + 8 * half];
            d2[i] = dv * dv;
        }
#pragma unroll
        for (int nt = 0; nt < 8; ++nt) {
            const int   n    = nt * 16 + r;
            const float bcol = bias[n];
            float* hp = h   + (size_t)m0 * HID + n;
            float* op = out + (size_t)m0 * HID + n;
#pragma unroll
            for (int i = 0; i < 8; ++i) {
                const size_t off = (size_t)(i + 8 * half) * HID;   // row m0 + i + 8*half
                const float  v   = acc[nt][i];
                hp[off] = v;
                op[off] = bcol + d2[i] * v;
            }
        }
    }
}

// ---------------- edge scatter: out[dst] += h[src] * dinv[src]*dinv[dst] ----------------
// 32 lanes per edge, 4 channels per lane (float4 gather, 4 fp32 atomics)
__global__ __launch_bounds__(256) void k_scatter(
    const int* __restrict__ ei, const float* __restrict__ h,
    const float* __restrict__ dinv, float* __restrict__ out, int E) {
    const long long gid = (long long)blockIdx.x * blockDim.x + threadIdx.x;
    const int e    = (int)(gid >> 5);
    const int lane = (int)(gid & 31);
    if (e >= E) return;
    const int src = ei[e];               // row (source j)
    const int dst = ei[(size_t)E + e];   // col (target i)
    const float nrm = dinv[src] * dinv[dst];
    const float4 hv = ((const float4*)(h + (size_t)src * HID))[lane];
    float* op = out + (size_t)dst * HID + 4 * lane;
    atomicAdd(op + 0, hv.x * nrm);
    atomicAdd(op + 1, hv.y * nrm);
    atomicAdd(op + 2, hv.z * nrm);
    atomicAdd(op + 3, hv.w * nrm);
}

// ---------------- PReLU in place ----------------
__global__ void k_prelu(float* __restrict__ out, const float* __restrict__ alpha, long long total) {
    long long i = (long long)blockIdx.x * blockDim.x + threadIdx.x;
    if (i < total) {
        float v = out[i];
        float a = alpha[(int)(i & (HID - 1))];
        out[i] = v > 0.0f ? v : a * v;
    }
}

extern "C" void kernel_launch(void* const* d_in, const int* in_sizes, int n_in,
                              void* d_out, int out_size, void* d_ws, size_t ws_size,
                              hipStream_t stream) {
    const float* x     = (const float*)d_in[0];
    const int*   ei    = (const int*)d_in[1];
    const float* W     = (const float*)d_in[2];
    const float* bias  = (const float*)d_in[3];
    const float* alpha = (const float*)d_in[4];
    float*       out   = (float*)d_out;

    const int N = in_sizes[0] / IN_CH;   // 100000
    const int E = in_sizes[1] / 2;       // 1600000

    // workspace: h [N*HID f32] then deg/dinv [N f32]
    float* h    = (float*)d_ws;
    float* dinv = h + (size_t)N * HID;

    k_deg_init <<<(N + 255) / 256, 256, 0, stream>>>(dinv, N);
    k_deg_edges<<<(E + 255) / 256, 256, 0, stream>>>(ei, dinv, E);
    k_dinv     <<<(N + 255) / 256, 256, 0, stream>>>(dinv, N);

    const int strips = (N + 15) / 16;        // 6250
    const int blocks = (strips + 7) / 8;     // 782
    k_gemm_wmma<<<blocks, 256, 0, stream>>>(x, W, bias, dinv, h, out, N);

    const long long scThreads = (long long)E * 32;
    k_scatter<<<(int)((scThreads + 255) / 256), 256, 0, stream>>>(ei, h, dinv, out, E);

    const long long total = (long long)N * HID;
    k_prelu<<<(int)((total + 255) / 256), 256, 0, stream>>>(out, alpha, total);
}